// HyperNode_30623116821338
// MI455X (gfx1250) — compile-verified
//
#include <hip/hip_runtime.h>
#include <hip/hip_bf16.h>

typedef float v2f __attribute__((ext_vector_type(2)));
typedef float v8f __attribute__((ext_vector_type(8)));

#define N_E0_C 100000
#define N_E1_C 50000
#define SLOPE 0.2f
#define BETA 0.5f
#define BN_EPS 1e-5f

// ---------------------------------------------------------------------------
// GEMM: Y[N,128] = X[N,128] @ W[128,128], fp32 via V_WMMA_F32_16X16X4_F32.
// Block: 256 thr = 8 waves; wave w computes rows [blk*128 + w*16, +16) x all 128 cols.
// W staged transposed in LDS with XOR swizzle -> conflict-free ds_load_b64 B frags.
// ---------------------------------------------------------------------------
#define SWZ(c) (((c) & 15) << 2)

__global__ __launch_bounds__(256) void gemm128_wmma(const float* __restrict__ X,
                                                    const float* __restrict__ W,
                                                    float* __restrict__ Y, int N)
{
    __shared__ float Wt[128 * 128];           // 64 KB, Wt[c][k ^ SWZ(c)] = W[k][c]
    const int tid = threadIdx.x;
    for (int i = tid; i < 128 * 128; i += 256) {
        int k = i >> 7, c = i & 127;
        Wt[c * 128 + (k ^ SWZ(c))] = W[i];
    }
    __syncthreads();

    const int lane = tid & 31;
    const int lm   = lane & 15;               // M (A) / N (B,D) index within tile
    const int hi   = lane >> 4;               // lane half selects K pair / row half
    const int row0 = blockIdx.x * 128 + (tid >> 5) * 16;
    if (row0 + 16 > N) return;                // wave-uniform: EXEC stays all-1s for WMMA

    // A fragments: lane holds X[row0+lm][4*kk + 2*hi + {0,1}] for kk = 0..31
    v2f a[32];
    const float* xp = X + (size_t)(row0 + lm) * 128 + 2 * hi;
#pragma unroll
    for (int kk = 0; kk < 32; ++kk)
        a[kk] = *(const v2f*)(xp + kk * 4);

    for (int ct = 0; ct < 8; ++ct) {
        const int col = ct * 16 + lm;
        const float* wp = Wt + col * 128;
        const int sw = SWZ(col);
        v8f acc = {0.f, 0.f, 0.f, 0.f, 0.f, 0.f, 0.f, 0.f};
#pragma unroll
        for (int kk = 0; kk < 32; ++kk) {
            v2f b = *(const v2f*)(wp + ((kk * 4 + 2 * hi) ^ sw));
            acc = __builtin_amdgcn_wmma_f32_16x16x4_f32(false, a[kk], false, b,
                                                        (short)0, acc, false, false);
        }
        float* yp = Y + (size_t)(row0 + hi * 8) * 128 + col;
#pragma unroll
        for (int r = 0; r < 8; ++r)
            yp[(size_t)r * 128] = acc[r];     // D: VGPR r -> rows row0+8*hi+r
    }
}

// ---------------------------------------------------------------------------
// Tiny 4x128 projection of edge-type/attr embeddings: tbl = emb(4x128) @ W
// ---------------------------------------------------------------------------
__global__ __launch_bounds__(128) void emb_proj(const float* __restrict__ emb,
                                                const float* __restrict__ W,
                                                float* __restrict__ tbl)
{
    int c = threadIdx.x;
    for (int r = 0; r < 4; ++r) {
        float acc = 0.f;
        for (int k = 0; k < 128; ++k) acc += emb[r * 128 + k] * W[k * 128 + c];
        tbl[r * 128 + c] = acc;
    }
}

// float atomic max via int min/max trick; works with 0xFFFFFFFF sentinel init
__device__ __forceinline__ void atomicMaxF(float* addr, float v)
{
    if (v >= 0.0f) atomicMax((int*)addr, __float_as_int(v));
    else           atomicMin((unsigned int*)addr, __float_as_uint(v));
}

// ---------------------------------------------------------------------------
// Pass A: per (edge, head): logits = leakyrelu(q[dst].k_e / 4); atomic max -> m
// ---------------------------------------------------------------------------
__global__ __launch_bounds__(256) void edge_logits(const float* __restrict__ q,
                                                   const float* __restrict__ kAll,
                                                   const float* __restrict__ keTbl,
                                                   const int* __restrict__ src,
                                                   const int* __restrict__ dst,
                                                   const int* __restrict__ et,
                                                   float* __restrict__ logits,
                                                   float* __restrict__ m, int E)
{
    int t = blockIdx.x * 256 + threadIdx.x;
    if (t >= E * 8) return;
    int e = t >> 3, h = t & 7;
    int s = src[e], d = dst[e], ty = et[e];
    const float4* qp = (const float4*)(q     + (size_t)d  * 128 + h * 16);
    const float4* kp = (const float4*)(kAll  + (size_t)s  * 128 + h * 16);
    const float4* ep = (const float4*)(keTbl + (size_t)ty * 128 + h * 16);
    float acc = 0.f;
#pragma unroll
    for (int i = 0; i < 4; ++i) {
        float4 qv = qp[i], kv = kp[i], ev = ep[i];
        acc += qv.x * (kv.x + ev.x) + qv.y * (kv.y + ev.y)
             + qv.z * (kv.z + ev.z) + qv.w * (kv.w + ev.w);
    }
    acc *= 0.25f;                                     // 1/sqrt(d), d=16
    float lg = acc >= 0.f ? acc : SLOPE * acc;        // leaky relu
    logits[t] = lg;
    atomicMaxF(&m[(size_t)d * 8 + h], lg);
}

// ---------------------------------------------------------------------------
// Pass B: a = exp(logit - m[dst]); s[dst] += a; acc[dst] += a * v_e (16 floats)
// (softmax denominator folded into the final combine -> only two edge sweeps)
// ---------------------------------------------------------------------------
__global__ __launch_bounds__(256) void edge_accum(const float* __restrict__ vAll,
                                                  const float* __restrict__ veTbl,
                                                  const float* __restrict__ logits,
                                                  const float* __restrict__ m,
                                                  const int* __restrict__ src,
                                                  const int* __restrict__ dst,
                                                  const int* __restrict__ et,
                                                  float* __restrict__ sexp,
                                                  float* __restrict__ accum, int E)
{
    int t = blockIdx.x * 256 + threadIdx.x;
    if (t >= E * 8) return;
    int e = t >> 3, h = t & 7;
    int s = src[e], d = dst[e], ty = et[e];
    float a = __expf(logits[t] - m[(size_t)d * 8 + h]);
    atomicAdd(&sexp[(size_t)d * 8 + h], a);
    const float4* vp = (const float4*)(vAll  + (size_t)s  * 128 + h * 16);
    const float4* ep = (const float4*)(veTbl + (size_t)ty * 128 + h * 16);
    float* op = accum + (size_t)d * 128 + h * 16;
#pragma unroll
    for (int i = 0; i < 4; ++i) {
        float4 vv = vp[i], ev = ep[i];
        atomicAdd(op + i * 4 + 0, a * (vv.x + ev.x));
        atomicAdd(op + i * 4 + 1, a * (vv.y + ev.y));
        atomicAdd(op + i * 4 + 2, a * (vv.z + ev.z));
        atomicAdd(op + i * 4 + 3, a * (vv.w + ev.w));
    }
}

// out = BETA*skip + (1-BETA) * acc / (s + 1e-16)
__global__ __launch_bounds__(256) void combine(const float* __restrict__ skip,
                                               const float* __restrict__ accum,
                                               const float* __restrict__ sexp,
                                               float* __restrict__ out, int N)
{
    int t = blockIdx.x * 256 + threadIdx.x;
    if (t >= N * 128) return;
    int i = t >> 7, c = t & 127, h = c >> 4;
    out[t] = BETA * skip[t]
           + (1.0f - BETA) * accum[t] / (sexp[(size_t)i * 8 + h] + 1e-16f);
}

// BatchNorm partial sums: 64 rows per block, per-column block reduction
__global__ __launch_bounds__(256) void bn_partial(const float* __restrict__ x,
                                                  float* __restrict__ gsum,
                                                  float* __restrict__ gsq, int N)
{
    __shared__ float ls[256], lq[256];
    int c = threadIdx.x & 127, half = threadIdx.x >> 7;
    int r0 = blockIdx.x * 64;
    int rend = r0 + 64; if (rend > N) rend = N;
    float s = 0.f, q = 0.f;
    for (int r = r0 + half; r < rend; r += 2) {
        float v = x[(size_t)r * 128 + c];
        s += v; q += v * v;
    }
    ls[threadIdx.x] = s; lq[threadIdx.x] = q;
    __syncthreads();
    if (half == 0) {
        atomicAdd(&gsum[c], ls[c] + ls[c + 128]);
        atomicAdd(&gsq[c],  lq[c] + lq[c + 128]);
    }
}

__global__ __launch_bounds__(256) void bn_apply(const float* __restrict__ xin,
                                                const float* __restrict__ gsum,
                                                const float* __restrict__ gsq,
                                                const float* __restrict__ gamma,
                                                const float* __restrict__ beta,
                                                float* __restrict__ xout, int N)
{
    int t = blockIdx.x * 256 + threadIdx.x;
    if (t >= N * 128) return;
    int c = t & 127;
    float inv = 1.0f / (float)N;
    float mu  = gsum[c] * inv;
    float var = gsq[c] * inv - mu * mu;
    xout[t] = (xin[t] - mu) * rsqrtf(var + BN_EPS) * gamma[c] + beta[c];
}

// ---------------------------------------------------------------------------
extern "C" void kernel_launch(void* const* d_in, const int* in_sizes, int n_in,
                              void* d_out, int out_size, void* d_ws, size_t ws_size,
                              hipStream_t stream)
{
    (void)n_in; (void)out_size; (void)ws_size;
    const float* x        = (const float*)d_in[0];
    const int*   ei0      = (const int*)d_in[1];
    const int*   et0      = (const int*)d_in[2];
    const int*   ei1      = (const int*)d_in[3];
    const int*   et1      = (const int*)d_in[4];
    const float* emb_type = (const float*)d_in[5];
    const float* emb_attr = (const float*)d_in[6];
    const float* Wq0 = (const float*)d_in[7],  *Wk0 = (const float*)d_in[8];
    const float* Wv0 = (const float*)d_in[9],  *We0 = (const float*)d_in[10];
    const float* Ws0 = (const float*)d_in[11];
    const float* Wq1 = (const float*)d_in[12], *Wk1 = (const float*)d_in[13];
    const float* Wv1 = (const float*)d_in[14], *We1 = (const float*)d_in[15];
    const float* Ws1 = (const float*)d_in[16];
    const float* bng = (const float*)d_in[17], *bnb = (const float*)d_in[18];

    const int NSRC = in_sizes[0] / 128;       // 200000
    const int E0   = in_sizes[1] / 2;         // 1000000
    const int E1   = in_sizes[3] / 2;         // 500000
    const int NE0  = N_E0_C, NE1 = N_E1_C;

    // workspace layout (reused across layers)
    char* w = (char*)d_ws; size_t off = 0;
    auto alloc = [&](size_t bytes) -> float* {
        float* p = (float*)(w + off);
        off = (off + bytes + 255) & ~(size_t)255;
        return p;
    };
    float* kAll   = alloc((size_t)NSRC * 128 * 4);
    float* vAll   = alloc((size_t)NSRC * 128 * 4);
    float* qb     = alloc((size_t)NE0 * 128 * 4);   // also reused as raw layer-0 out
    float* skip   = alloc((size_t)NE0 * 128 * 4);
    float* logits = alloc((size_t)E0 * 8 * 4);
    float* mbuf   = alloc((size_t)NE0 * 8 * 4);
    float* sbuf   = alloc((size_t)NE0 * 8 * 4);
    float* accum  = alloc((size_t)NE0 * 128 * 4);
    float* hbuf   = alloc((size_t)NE0 * 128 * 4);   // BN output, layer-1 source
    float* keT    = alloc(4 * 128 * 4);
    float* veT    = alloc(4 * 128 * 4);
    float* gsum   = alloc(128 * 4);
    float* gsq    = alloc(128 * 4);

    const int*  src0 = ei0;       const int* dst0 = ei0 + E0;
    const int*  src1 = ei1;      const int* dst1 = ei1 + E1;

    // ---------------- layer 0: v2e ----------------
    int gN  = (NSRC + 127) / 128;             // 1563
    int gD0 = (NE0 + 127) / 128;              // 782
    gemm128_wmma<<<gN,  256, 0, stream>>>(x, Wk0, kAll, NSRC);
    gemm128_wmma<<<gN,  256, 0, stream>>>(x, Wv0, vAll, NSRC);
    gemm128_wmma<<<gD0, 256, 0, stream>>>(x, Wq0, qb, NE0);
    gemm128_wmma<<<gD0, 256, 0, stream>>>(x, Ws0, skip, NE0);
    emb_proj<<<1, 128, 0, stream>>>(emb_type, We0, keT);
    emb_proj<<<1, 128, 0, stream>>>(emb_attr, We0, veT);

    hipMemsetAsync(mbuf,  0xFF, (size_t)NE0 * 8 * 4,   stream);   // -inf-like sentinel
    hipMemsetAsync(sbuf,  0,    (size_t)NE0 * 8 * 4,   stream);
    hipMemsetAsync(accum, 0,    (size_t)NE0 * 128 * 4, stream);

    int gE0 = (E0 * 8 + 255) / 256;
    edge_logits<<<gE0, 256, 0, stream>>>(qb, kAll, keT, src0, dst0, et0, logits, mbuf, E0);
    edge_accum <<<gE0, 256, 0, stream>>>(vAll, veT, logits, mbuf, src0, dst0, et0, sbuf, accum, E0);
    combine<<<(NE0 * 128 + 255) / 256, 256, 0, stream>>>(skip, accum, sbuf, qb, NE0);

    hipMemsetAsync(gsum, 0, 128 * 4, stream);
    hipMemsetAsync(gsq,  0, 128 * 4, stream);
    bn_partial<<<(NE0 + 63) / 64, 256, 0, stream>>>(qb, gsum, gsq, NE0);
    bn_apply<<<(NE0 * 128 + 255) / 256, 256, 0, stream>>>(qb, gsum, gsq, bng, bnb, hbuf, NE0);

    // ---------------- layer 1: e2rel ----------------
    int gH  = (NE0 + 127) / 128;              // 782
    int gD1 = (NE1 + 127) / 128;              // 391
    gemm128_wmma<<<gH,  256, 0, stream>>>(hbuf, Wk1, kAll, NE0);
    gemm128_wmma<<<gH,  256, 0, stream>>>(hbuf, Wv1, vAll, NE0);
    gemm128_wmma<<<gD1, 256, 0, stream>>>(hbuf, Wq1, qb, NE1);
    gemm128_wmma<<<gD1, 256, 0, stream>>>(hbuf, Ws1, skip, NE1);
    emb_proj<<<1, 128, 0, stream>>>(emb_type, We1, keT);
    emb_proj<<<1, 128, 0, stream>>>(emb_attr, We1, veT);

    hipMemsetAsync(mbuf,  0xFF, (size_t)NE1 * 8 * 4,   stream);
    hipMemsetAsync(sbuf,  0,    (size_t)NE1 * 8 * 4,   stream);
    hipMemsetAsync(accum, 0,    (size_t)NE1 * 128 * 4, stream);

    int gE1 = (E1 * 8 + 255) / 256;
    edge_logits<<<gE1, 256, 0, stream>>>(qb, kAll, keT, src1, dst1, et1, logits, mbuf, E1);
    edge_accum <<<gE1, 256, 0, stream>>>(vAll, veT, logits, mbuf, src1, dst1, et1, sbuf, accum, E1);
    combine<<<(NE1 * 128 + 255) / 256, 256, 0, stream>>>(skip, accum, sbuf, (float*)d_out, NE1);
}